// NormBatched_8443905703957
// MI455X (gfx1250) — compile-verified
//
#include <hip/hip_runtime.h>

typedef __attribute__((ext_vector_type(2))) float v2f;
typedef __attribute__((ext_vector_type(8))) float v8f;

#define C_DIM   256
#define S_DIM   4096      // H*W
#define B_DIM   32
#define NT      16        // channel tiles of 16
#define NCHUNK  8         // spatial chunks (one per wave)
#define AFWD    0.999f
#define EPS     1e-5f

// ---------------- Kernel A: per-(b, channel) Σx and Σx² via WMMA ----------------
// Block = 256 threads = 8 waves. blockIdx.x = b*16 + tile.
// Wave w reduces spatial slice [w*512, w*512+512) for its 16-channel tile using
// V_WMMA_F32_16X16X4_F32 with A = x-values (M=channel, K=spatial), B = ones.
// D[m][*] then holds the channel-m partial sum, extracted from lanes 0 and 16.
__global__ __launch_bounds__(256) void onorm_partials_kernel(
    const float* __restrict__ x, float* __restrict__ psum, float* __restrict__ psq) {
  const int blk  = blockIdx.x;
  const int b    = blk >> 4;
  const int tile = blk & 15;
  const int wave = threadIdx.x >> 5;
  const int lane = threadIdx.x & 31;

  const int ch    = (tile << 4) + (lane & 15);
  const int sbase = (wave << 9) + ((lane >> 4) << 1);   // wave*512 + {0,2}

  const float* xb = x + (size_t)b * S_DIM * C_DIM + ch;

  v8f acc  = {};
  v8f accq = {};
  v2f bones = {1.0f, 1.0f};

  for (int i = 0; i < 128; ++i) {
    const size_t s = (size_t)(sbase + (i << 2));
    float a0 = xb[s * C_DIM];            // (s,   ch)
    float a1 = xb[(s + 1) * C_DIM];      // (s+1, ch)
    v2f a   = {a0, a1};
    v2f asq = {a0 * a0, a1 * a1};
    // D = A(16x4) * ones(4x16) + C  => row m of D = channel-m partial sum
    acc  = __builtin_amdgcn_wmma_f32_16x16x4_f32(false, a,   false, bones,
                                                 (short)0, acc,  false, false);
    accq = __builtin_amdgcn_wmma_f32_16x16x4_f32(false, asq, false, bones,
                                                 (short)0, accq, false, false);
  }

  // Lane 0 holds D[0..7][0] in acc[0..7]; lane 16 holds D[8..15][0].
  if ((lane & 15) == 0) {
    const int chbase = (lane >> 4) << 3;  // 0 or 8
    const size_t base = ((((size_t)b * NT + tile) * NCHUNK + wave) * 16) + chbase;
    float* ps = psum + base;
    float* pq = psq  + base;
#pragma unroll
    for (int r = 0; r < 8; ++r) {
      ps[r] = acc[r];
      pq[r] = accq[r];
    }
  }
}

// ---------------- Kernel B: 32-step streaming-stat recurrence ----------------
// One block, 256 threads (one per channel). Reduces the 8 spatial partials per
// (b, c), then runs the EMA chain, emitting mu_{t-1} and 1/sqrt(var_{t-1}+eps).
__global__ __launch_bounds__(256) void onorm_recurrence_kernel(
    const float* __restrict__ mu0, const float* __restrict__ var0,
    const float* __restrict__ psum, const float* __restrict__ psq,
    float* __restrict__ muprev, float* __restrict__ invs) {
  const int c    = threadIdx.x;
  const int tile = c >> 4;
  const int ci   = c & 15;

  float mu  = mu0[c];
  float var = var0[c];

  for (int b = 0; b < B_DIM; ++b) {
    float s1 = 0.0f, s2 = 0.0f;
#pragma unroll
    for (int k = 0; k < NCHUNK; ++k) {
      const size_t base = (((size_t)b * NT + tile) * NCHUNK + k) * 16 + ci;
      s1 += psum[base];
      s2 += psq[base];
    }
    const float m = s1 * (1.0f / (float)S_DIM);
    float v = s2 * (1.0f / (float)S_DIM) - m * m;
    v = fmaxf(v, 0.0f);

    muprev[b * C_DIM + c] = mu;
    invs[b * C_DIM + c]   = 1.0f / sqrtf(var + EPS);

    const float d = m - mu;
    var = AFWD * var + (1.0f - AFWD) * v + AFWD * (1.0f - AFWD) * d * d;
    mu  = mu + (1.0f - AFWD) * d;
  }
}

// ---------------- Kernel C: normalize (bandwidth kernel) ----------------
// x is 134 MB < 192 MB L2, so this re-read is L2-serviced; the write is the
// only new HBM traffic. Each thread owns one channel-quad of one sample:
// stats loaded once into registers, then 16 strided float4 elementwise ops.
__global__ __launch_bounds__(256) void onorm_apply_kernel(
    const float* __restrict__ x, const float* __restrict__ muprev,
    const float* __restrict__ invs, float* __restrict__ out) {
  const int b     = blockIdx.x >> 6;
  const int strip = blockIdx.x & 63;
  const int c4    = threadIdx.x & 63;               // channel quad 0..63
  const int s0    = (strip << 6) + (threadIdx.x >> 6);

  const float4* xb = (const float4*)x   + (size_t)b * S_DIM * 64 + c4;
  float4*       ob = (float4*)out       + (size_t)b * S_DIM * 64 + c4;
  const float4  m4 = ((const float4*)muprev)[b * 64 + c4];
  const float4  i4 = ((const float4*)invs)[b * 64 + c4];

#pragma unroll
  for (int i = 0; i < 16; ++i) {
    const size_t off = (size_t)(s0 + (i << 2)) * 64;
    float4 xv = xb[off];
    float4 o;
    o.x = (xv.x - m4.x) * i4.x;
    o.y = (xv.y - m4.y) * i4.y;
    o.z = (xv.z - m4.z) * i4.z;
    o.w = (xv.w - m4.w) * i4.w;
    ob[off] = o;
  }
}

extern "C" void kernel_launch(void* const* d_in, const int* in_sizes, int n_in,
                              void* d_out, int out_size, void* d_ws, size_t ws_size,
                              hipStream_t stream) {
  const float* x    = (const float*)d_in[0];
  const float* mu0  = (const float*)d_in[1];
  const float* var0 = (const float*)d_in[2];
  // d_in[3] (u0), d_in[4] (v0) are backward-only controls; unused in forward.
  float* out = (float*)d_out;

  float* ws     = (float*)d_ws;
  float* psum   = ws;                    // 32*16*8*16 = 65536 floats
  float* psq    = ws + 65536;            // 65536 floats
  float* muprev = ws + 131072;           // 32*256 = 8192 floats
  float* invs   = ws + 139264;           // 8192 floats  (total 576 KB)

  onorm_partials_kernel<<<B_DIM * NT, 256, 0, stream>>>(x, psum, psq);
  onorm_recurrence_kernel<<<1, 256, 0, stream>>>(mu0, var0, psum, psq, muprev, invs);
  onorm_apply_kernel<<<B_DIM * 64, 256, 0, stream>>>(x, muprev, invs, out);
}